// USVBlock_36077725287158
// MI455X (gfx1250) — compile-verified
//
#include <hip/hip_runtime.h>
#include <hip/hip_bf16.h>

// Problem dims (fixed by the reference)
constexpr int B = 4, N = 2048, M = 128, D = 64;
constexpr int KSPLIT = 8;                 // K-partitions of aggregation GEMM
constexpr int NTILES = B * (M / 16);      // 32 output tiles
constexpr float NEG_SLOPE = 0.2f;

typedef __attribute__((ext_vector_type(2))) float v2f;
typedef __attribute__((ext_vector_type(8))) float v8f;

__device__ inline v8f wmma_f32_4(v2f a, v2f b, v8f c) {
    // D(16x16,f32) = A(16x4,f32) x B(4x16,f32) + C
    return __builtin_amdgcn_wmma_f32_16x16x4_f32(
        /*neg_a=*/false, a, /*neg_b=*/false, b,
        /*c_mod=*/(short)0, c, /*reuse_a=*/false, /*reuse_b=*/false);
}

__device__ inline float lrelu(float x) { return x > 0.f ? x : NEG_SLOPE * x; }

// ---------------------------------------------------------------------------
// Kernel 1: u = W_dst^T attn_r (shared), v = W_src^T attn_l (ws), c_e scalar,
//           er[b,m] = feat_dst[b,m,:] . u          (one block, 512 threads)
// ---------------------------------------------------------------------------
__global__ void prep_kernel(const float* __restrict__ feat_dst,
                            const float* __restrict__ W_src,
                            const float* __restrict__ W_dst,
                            const float* __restrict__ W_edge,
                            const float* __restrict__ attn_l,
                            const float* __restrict__ attn_r,
                            const float* __restrict__ attn_e,
                            float* __restrict__ er,
                            float* __restrict__ v_out,
                            float* __restrict__ ce_out) {
    __shared__ float u[D];
    const int t = threadIdx.x;
    if (t < D) {
        float s = 0.f;
        for (int d = 0; d < D; ++d) s += W_dst[d * D + t] * attn_r[d];
        u[t] = s;
    } else if (t < 2 * D) {
        const int k = t - D;
        float s = 0.f;
        for (int d = 0; d < D; ++d) s += W_src[d * D + k] * attn_l[d];
        v_out[k] = s;
    } else if (t == 2 * D) {
        float s = 0.f;
        for (int d = 0; d < D; ++d) s += W_edge[d] * attn_e[d];
        ce_out[0] = s;
    }
    __syncthreads();
    // t indexes the flat (b,m) pair, B*M == 512 == blockDim.x
    const float* f = feat_dst + (size_t)t * D;
    float s = 0.f;
    for (int k = 0; k < D; ++k) s += f[k] * u[k];
    er[t] = s;
}

// ---------------------------------------------------------------------------
// Kernel 2: el[b,n] = feat_src[b,n,:] . v
// ---------------------------------------------------------------------------
__global__ void el_kernel(const float* __restrict__ feat_src,
                          const float* __restrict__ v,
                          float* __restrict__ el) {
    const int i = blockIdx.x * blockDim.x + threadIdx.x;  // < B*N
    const float* f = feat_src + (size_t)i * D;
    float s = 0.f;
    for (int k = 0; k < D; ++k) s += f[k] * v[k];
    el[i] = s;
}

// ---------------------------------------------------------------------------
// Kernel 3: hs = feat_src @ W_src^T  via V_WMMA_F32_16X16X4_F32
//           one wave per 16-row n-tile; 4 waves per block.
// ---------------------------------------------------------------------------
__global__ void hs_gemm_kernel(const float* __restrict__ feat_src,
                               const float* __restrict__ W_src,
                               float* __restrict__ hs) {
    const int wave = threadIdx.x >> 5;
    const int lane = threadIdx.x & 31;
    const int lo = lane & 15;
    const int hi = lane >> 4;
    const int tile = blockIdx.x * 4 + wave;        // 0 .. B*N/16-1
    const int b  = tile / (N / 16);
    const int n0 = (tile % (N / 16)) * 16;

    // A[mi][k] = feat_src[b][n0+mi][k]; lane holds mi=lo, k = k0 + hi*2 + {0,1}
    const float* A0 = feat_src + ((size_t)b * N + n0 + lo) * D + hi * 2;
    v8f acc[4] = {v8f{}, v8f{}, v8f{}, v8f{}};
    for (int k0 = 0; k0 < D; k0 += 4) {
        v2f a;
        a.x = A0[k0];
        a.y = A0[k0 + 1];
#pragma unroll
        for (int t = 0; t < 4; ++t) {
            // B[k][d] = W_src[d][k]; lane holds d = t*16+lo, k = k0 + hi*2 + {0,1}
            const float* Bp = W_src + (size_t)(t * 16 + lo) * D + k0 + hi * 2;
            v2f bb;
            bb.x = Bp[0];
            bb.y = Bp[1];
            acc[t] = wmma_f32_4(a, bb, acc[t]);
        }
    }
    // D tile: VGPR j -> row (j + hi*8), col lo (per d-tile t)
    float* H = hs + ((size_t)b * N + n0) * D;
#pragma unroll
    for (int t = 0; t < 4; ++t)
#pragma unroll
        for (int j = 0; j < 8; ++j)
            H[(size_t)(j + hi * 8) * D + t * 16 + lo] = acc[t][j];
}

// ---------------------------------------------------------------------------
// Kernel 4: single-pass ONLINE softmax stats per (b,m):
//           rmax, 1/sum(exp), s_e = sum(alpha*feat_edge)
// ---------------------------------------------------------------------------
__global__ void stats_kernel(const float* __restrict__ fe,
                             const int* __restrict__ adj,
                             const float* __restrict__ el,
                             const float* __restrict__ er,
                             const float* __restrict__ ce_p,
                             float* __restrict__ rmax,
                             float* __restrict__ rrsum,
                             float* __restrict__ se_out) {
    const int bm = blockIdx.x;
    const int b = bm / M, m = bm % M;
    const float ce = ce_p[0], erm = er[bm];
    __shared__ float smx[256];
    __shared__ float ssm[256];
    __shared__ float sse[256];

    float mx = -3.0e38f, sm = 0.f, se = 0.f;
    for (int n = threadIdx.x; n < N; n += 256) {
        const size_t idx = ((size_t)b * N + n) * M + m;
        if (adj[idx]) {
            const float fev = fe[idx];
            const float e = lrelu(el[b * N + n] + ce * fev + erm);
            if (e > mx) {                       // rescale running sums
                const float sc = __expf(mx - e);
                sm *= sc;
                se *= sc;
                mx = e;
            }
            const float p = __expf(e - mx);
            sm += p;
            se += p * fev;
        }
    }
    smx[threadIdx.x] = mx;
    ssm[threadIdx.x] = sm;
    sse[threadIdx.x] = se;
    __syncthreads();
    for (int s = 128; s > 0; s >>= 1) {
        if (threadIdx.x < s) {
            const float m1 = smx[threadIdx.x], m2 = smx[threadIdx.x + s];
            const float mm = fmaxf(m1, m2);
            const float c1 = __expf(m1 - mm), c2 = __expf(m2 - mm);
            smx[threadIdx.x] = mm;
            ssm[threadIdx.x] = ssm[threadIdx.x] * c1 + ssm[threadIdx.x + s] * c2;
            sse[threadIdx.x] = sse[threadIdx.x] * c1 + sse[threadIdx.x + s] * c2;
        }
        __syncthreads();
    }
    if (threadIdx.x == 0) {
        rmax[bm]   = smx[0];
        rrsum[bm]  = 1.f / ssm[0];
        se_out[bm] = sse[0] / ssm[0];
    }
}

// ---------------------------------------------------------------------------
// Kernel 5: K-split partial aggregation: partial[kc][tile] = alpha^T @ hs
//           over n in [kc*256, kc*256+256).  Grid (NTILES, KSPLIT), 4 waves.
//           Deterministic: each block owns its partial tile (no atomics).
// ---------------------------------------------------------------------------
__global__ void aggregate_partial_kernel(const float* __restrict__ fe,
                                         const int* __restrict__ adj,
                                         const float* __restrict__ hs,
                                         const float* __restrict__ el,
                                         const float* __restrict__ er,
                                         const float* __restrict__ rmax,
                                         const float* __restrict__ rrsum,
                                         const float* __restrict__ ce_p,
                                         float* __restrict__ partial) {
    __shared__ float lds[4 * 16 * D];
    const int wave = threadIdx.x >> 5;
    const int lane = threadIdx.x & 31;
    const int lo = lane & 15;
    const int hi = lane >> 4;
    const int tile = blockIdx.x;               // 0 .. NTILES-1
    const int kc   = blockIdx.y;               // 0 .. KSPLIT-1
    const int b  = tile / (M / 16);
    const int m0 = (tile % (M / 16)) * 16;
    const int m  = m0 + lo;

    const float ce  = ce_p[0];
    const float erm = er[b * M + m];
    const float mxm = rmax[b * M + m];
    const float rsm = rrsum[b * M + m];

    v8f acc[4] = {v8f{}, v8f{}, v8f{}, v8f{}};
    const int kspan = N / (KSPLIT * 4);        // 64 source nodes per wave
    const int nbeg  = kc * (N / KSPLIT) + wave * kspan;
    for (int n0 = nbeg; n0 < nbeg + kspan; n0 += 4) {
        const int n = n0 + hi * 2;
        // A[mi][k] = alpha[n0+k][m0+mi]; lane holds mi=lo, k = hi*2 + {0,1}
        const size_t idx0 = ((size_t)b * N + n) * M + m;
        v2f a;
        {
            const int   ad0 = adj[idx0];
            const float e0  = lrelu(el[b * N + n] + ce * fe[idx0] + erm);
            a.x = ad0 ? __expf(e0 - mxm) * rsm : 0.f;
            const int   ad1 = adj[idx0 + M];
            const float e1  = lrelu(el[b * N + n + 1] + ce * fe[idx0 + M] + erm);
            a.y = ad1 ? __expf(e1 - mxm) * rsm : 0.f;
        }
        // B[k][d] = hs[b][n0+k][d]; lane holds d = t*16+lo, k = hi*2 + {0,1}
        const float* H = hs + ((size_t)b * N + n) * D;
#pragma unroll
        for (int t = 0; t < 4; ++t) {
            v2f bb;
            bb.x = H[t * 16 + lo];
            bb.y = H[D + t * 16 + lo];
            acc[t] = wmma_f32_4(a, bb, acc[t]);
        }
    }
    // spill per-wave partial D tiles: row mi = j + hi*8, col d = t*16+lo
#pragma unroll
    for (int t = 0; t < 4; ++t)
#pragma unroll
        for (int j = 0; j < 8; ++j)
            lds[wave * (16 * D) + (j + hi * 8) * D + t * 16 + lo] = acc[t][j];
    __syncthreads();

    // reduce 4 waves -> this block's partial tile
    float* P = partial + ((size_t)kc * NTILES + tile) * (16 * D);
    for (int idx = threadIdx.x; idx < 16 * D; idx += blockDim.x)
        P[idx] = lds[idx] + lds[16 * D + idx] + lds[2 * 16 * D + idx] +
                 lds[3 * 16 * D + idx];
}

// ---------------------------------------------------------------------------
// Kernel 6: finalize: sum KSPLIT partials + W_edge (x) s_e, sigmoid -> out
// ---------------------------------------------------------------------------
__global__ void finalize_kernel(const float* __restrict__ partial,
                                const float* __restrict__ se,
                                const float* __restrict__ W_edge,
                                float* __restrict__ out) {
    const int g = blockIdx.x * blockDim.x + threadIdx.x;  // < B*M*D
    const int b = g / (M * D);
    const int r = g % (M * D);
    const int m = r >> 6;                 // / D
    const int d = r & (D - 1);
    const int tile = b * (M / 16) + (m >> 4);
    const int mi   = m & 15;
    const size_t poff = (size_t)tile * (16 * D) + mi * D + d;

    float s = 0.f;
#pragma unroll
    for (int c = 0; c < KSPLIT; ++c)
        s += partial[(size_t)c * NTILES * (16 * D) + poff];
    s += W_edge[d] * se[b * M + m];
    out[g] = 1.f / (1.f + __expf(-s));
}

// ---------------------------------------------------------------------------
// Workspace layout (floats)
// ---------------------------------------------------------------------------
constexpr size_t WS_HS      = 0;                          // B*N*D = 524288
constexpr size_t WS_EL      = WS_HS + (size_t)B * N * D;  // B*N   = 8192
constexpr size_t WS_ER      = WS_EL + (size_t)B * N;      // B*M   = 512
constexpr size_t WS_RMAX    = WS_ER + (size_t)B * M;
constexpr size_t WS_RRSUM   = WS_RMAX + (size_t)B * M;
constexpr size_t WS_SE      = WS_RRSUM + (size_t)B * M;
constexpr size_t WS_V       = WS_SE + (size_t)B * M;      // D
constexpr size_t WS_CE      = WS_V + D;                   // 1
constexpr size_t WS_PARTIAL = WS_CE + 63;                 // align-ish; KSPLIT*NTILES*16*D = 262144

extern "C" void kernel_launch(void* const* d_in, const int* in_sizes, int n_in,
                              void* d_out, int out_size, void* d_ws, size_t ws_size,
                              hipStream_t stream) {
    (void)in_sizes; (void)n_in; (void)out_size; (void)ws_size;
    const float* feat_src  = (const float*)d_in[0];
    const float* feat_dst  = (const float*)d_in[1];
    const float* feat_edge = (const float*)d_in[2];
    const int*   adj       = (const int*)d_in[3];
    const float* W_src     = (const float*)d_in[4];
    const float* W_dst     = (const float*)d_in[5];
    const float* W_edge    = (const float*)d_in[6];
    const float* attn_l    = (const float*)d_in[7];
    const float* attn_r    = (const float*)d_in[8];
    const float* attn_e    = (const float*)d_in[9];
    float* out = (float*)d_out;
    float* ws  = (float*)d_ws;

    float* hs      = ws + WS_HS;
    float* el      = ws + WS_EL;
    float* er      = ws + WS_ER;
    float* rmax    = ws + WS_RMAX;
    float* rrsum   = ws + WS_RRSUM;
    float* se      = ws + WS_SE;
    float* vvec    = ws + WS_V;
    float* cep     = ws + WS_CE;
    float* partial = ws + WS_PARTIAL;

    prep_kernel<<<1, B * M, 0, stream>>>(feat_dst, W_src, W_dst, W_edge,
                                         attn_l, attn_r, attn_e, er, vvec, cep);
    el_kernel<<<(B * N) / 256, 256, 0, stream>>>(feat_src, vvec, el);
    hs_gemm_kernel<<<(B * N / 16) / 4, 128, 0, stream>>>(feat_src, W_src, hs);
    stats_kernel<<<B * M, 256, 0, stream>>>(feat_edge, adj, el, er, cep,
                                            rmax, rrsum, se);
    aggregate_partial_kernel<<<dim3(NTILES, KSPLIT), 128, 0, stream>>>(
        feat_edge, adj, hs, el, er, rmax, rrsum, cep, partial);
    finalize_kernel<<<(B * M * D) / 256, 256, 0, stream>>>(partial, se, W_edge, out);
}